// PointNetPlusPlusSetAbstraction_15006615734064
// MI455X (gfx1250) — compile-verified
//
#include <hip/hip_runtime.h>

typedef __attribute__((ext_vector_type(2))) float v2f;
typedef __attribute__((ext_vector_type(8))) float v8f;

// ---------------- problem constants ----------------
#define PB   16      // batch
#define PN   16384   // points per cloud
#define PD   64      // input feature channels
#define PNP  1024    // sampled centroids
#define PNS  32      // group size
#define PM   (PB * PNP * PNS)  // 524288 GEMM rows
#define K0   67
#define K0P  68      // padded to multiple of 4
#define C0   64
#define C1   128
#define C2   128

// ---------------- FPS: one 1024-thread block per batch ----------------
#define FPS_THREADS 1024
#define FPS_PPT (PN / FPS_THREADS)   // 16 points per thread, kept in registers

__global__ __launch_bounds__(FPS_THREADS)
void fps_kernel(const float* __restrict__ xyz, const int* __restrict__ finit,
                int* __restrict__ cents)
{
    const int b = blockIdx.x;
    const int t = threadIdx.x;
    const float* xb = xyz + (size_t)b * 3 * PN;

    float px[FPS_PPT], py[FPS_PPT], pz[FPS_PPT], dist[FPS_PPT];
#pragma unroll
    for (int i = 0; i < FPS_PPT; ++i) {
        int n = t + i * FPS_THREADS;
        px[i] = xb[n];
        py[i] = xb[PN + n];
        pz[i] = xb[2 * PN + n];
        dist[i] = 1e10f;
    }

    __shared__ float sD[32];
    __shared__ int   sN[32];
    __shared__ int   sFar;

    int far = finit[b];
    for (int s = 0; s < PNP; ++s) {
        if (t == 0) cents[b * PNP + s] = far;
        float cx = xb[far], cy = xb[PN + far], cz = xb[2 * PN + far];

        float bd = -1.0f; int bn = 0;
#pragma unroll
        for (int i = 0; i < FPS_PPT; ++i) {
            float dx = px[i] - cx, dy = py[i] - cy, dz = pz[i] - cz;
            float d  = dx * dx + dy * dy + dz * dz;
            d = fminf(d, dist[i]);
            dist[i] = d;
            if (d > bd) { bd = d; bn = t + i * FPS_THREADS; }
        }
        // wave32 argmax
        for (int off = 16; off > 0; off >>= 1) {
            float od = __shfl_xor(bd, off);
            int   on = __shfl_xor(bn, off);
            if (od > bd) { bd = od; bn = on; }
        }
        if ((t & 31) == 0) { sD[t >> 5] = bd; sN[t >> 5] = bn; }
        __syncthreads();
        if (t < 32) {
            bd = sD[t]; bn = sN[t];
            for (int off = 16; off > 0; off >>= 1) {
                float od = __shfl_xor(bd, off);
                int   on = __shfl_xor(bn, off);
                if (od > bd) { bd = od; bn = on; }
            }
            if (t == 0) sFar = bn;
        }
        __syncthreads();
        far = sFar;
    }
}

// ---------------- new_xyz output: [B,3,NP] ----------------
__global__ void newxyz_kernel(const float* __restrict__ xyz,
                              const int* __restrict__ cents,
                              float* __restrict__ out)
{
    int i = blockIdx.x * blockDim.x + threadIdx.x;
    if (i >= PB * 3 * PNP) return;
    int s = i % PNP;
    int c = (i / PNP) % 3;
    int b = i / (3 * PNP);
    out[i] = xyz[((size_t)b * 3 + c) * PN + cents[b * PNP + s]];
}

// ---------------- pad W0 (64x67) -> (64x68) ----------------
__global__ void padw_kernel(const float* __restrict__ W0, float* __restrict__ Wp)
{
    int i = blockIdx.x * blockDim.x + threadIdx.x;
    if (i >= C0 * K0P) return;
    int k = i % K0P, n = i / K0P;
    Wp[i] = (k < K0) ? W0[n * K0 + k] : 0.0f;
}

// ---------------- gather+concat: X0 [M, 68] ----------------
__global__ void gather_kernel(const float* __restrict__ xyz,
                              const float* __restrict__ points,
                              const int* __restrict__ gidx,
                              float* __restrict__ X0)
{
    long i = (long)blockIdx.x * blockDim.x + threadIdx.x;
    if (i >= (long)PM * K0P) return;
    int  k = (int)(i % K0P);
    long m = i / K0P;
    int  b = (int)(m / (PNP * PNS));
    int  r = (int)(m % (PNP * PNS));
    int idx = gidx[(size_t)b * PNP * PNS + r];
    float v = 0.0f;
    if (k < 3)            v = xyz[((size_t)b * 3 + k) * PN + idx];
    else if (k < 3 + PD)  v = points[((size_t)b * PD + (k - 3)) * PN + idx];
    X0[i] = v;
}

// ---------------- WMMA f32 GEMM: Y[M,C] = f(X)[M,K] * W[C,K]^T ----------------
// Template-specialized: branchless inner loop, one wave computes a full 16xC
// output strip (C/16 accumulators), X is read from HBM exactly once, weights
// are staged in LDS once per block and B-fragments come from ds_load_b64.
// f(x) = NORM ? relu(x*scale[k]+shift[k]) : x  (BN+ReLU of previous layer
// fused into the A-fragment load; conv bias dropped: cancelled by BN mean).
template<int K, int C, bool NORM>
__global__ __launch_bounds__(256)
void gemm_wmma(const float* __restrict__ X,
               const float* __restrict__ W,
               float* __restrict__ Y,
               const float* __restrict__ scale,
               const float* __restrict__ shift)
{
    constexpr int NT = C / 16;
    __shared__ float sW[C * K];
    for (int i = threadIdx.x; i < C * K; i += 256) sW[i] = W[i];
    __syncthreads();

    const int wave = (blockIdx.x * blockDim.x + threadIdx.x) >> 5;  // m-tile
    const int lane = threadIdx.x & 31;
    const int l    = lane & 15;
    const int half = lane >> 4;

    const float* xrow  = X + (size_t)(wave * 16 + l) * K;
    const float* swrow = sW + l * K;

    v8f acc[NT];
#pragma unroll
    for (int t = 0; t < NT; ++t) acc[t] = {};

#pragma unroll 4
    for (int k = 0; k < K; k += 4) {
        const int kk = k + (half << 1);
        v2f a = *(const v2f*)(xrow + kk);
        if (NORM) {
            v2f sc = *(const v2f*)(scale + kk);
            v2f sh = *(const v2f*)(shift + kk);
            a.x = fmaxf(a.x * sc.x + sh.x, 0.0f);
            a.y = fmaxf(a.y * sc.y + sh.y, 0.0f);
        }
#pragma unroll
        for (int t = 0; t < NT; ++t) {
            v2f bf = *(const v2f*)(swrow + t * 16 * K + kk);
            acc[t] = __builtin_amdgcn_wmma_f32_16x16x4_f32(
                false, a, false, bf, (short)0, acc[t], false, false);
        }
    }

    float* yb = Y + (size_t)(wave * 16 + (half << 3)) * C + l;
#pragma unroll
    for (int t = 0; t < NT; ++t)
#pragma unroll
        for (int r = 0; r < 8; ++r)
            yb[(size_t)r * C + t * 16] = acc[t][r];
}

// ---------------- per-channel sum / sumsq over Y[M,C] ----------------
__global__ void zero_acc(float* a, float* b)
{
    int t = threadIdx.x;
    if (t < 128) { a[t] = 0.0f; b[t] = 0.0f; }
}

__global__ __launch_bounds__(256)
void stats_kernel(const float* __restrict__ Y, int C, int rowsPerBlock,
                  float* __restrict__ accS, float* __restrict__ accQ)
{
    __shared__ float ls[128], lq[128];
    int t = threadIdx.x;
    if (t < C) { ls[t] = 0.0f; lq[t] = 0.0f; }
    __syncthreads();

    int c     = t % C;
    int ro    = t / C;
    int rstep = blockDim.x / C;
    long r0   = (long)blockIdx.x * rowsPerBlock;

    float s = 0.0f, q = 0.0f;
    for (int r = ro; r < rowsPerBlock; r += rstep) {
        float v = Y[(size_t)(r0 + r) * C + c];
        s += v; q += v * v;
    }
    atomicAdd(&ls[c], s);
    atomicAdd(&lq[c], q);
    __syncthreads();
    if (t < C) {
        atomicAdd(&accS[t], ls[t]);
        atomicAdd(&accQ[t], lq[t]);
    }
}

// mean/var -> scale = g*rsqrt(var+eps), shift = be - mean*scale
__global__ void finalize_stats(const float* __restrict__ accS,
                               const float* __restrict__ accQ,
                               const float* __restrict__ g,
                               const float* __restrict__ be,
                               int C, float invM,
                               float* __restrict__ scale,
                               float* __restrict__ shift)
{
    int c = threadIdx.x;
    if (c >= C) return;
    float mean = accS[c] * invM;
    float var  = accQ[c] * invM - mean * mean;
    float sc   = g[c] * rsqrtf(var + 1e-5f);
    scale[c] = sc;
    shift[c] = be[c] - mean * sc;
}

// ---------------- final norm+relu+max over NS, transposed store ----------------
__global__ void maxpool_kernel(const float* __restrict__ Y,
                               const float* __restrict__ scale,
                               const float* __restrict__ shift,
                               float* __restrict__ out)
{
    int i = blockIdx.x * blockDim.x + threadIdx.x;
    if (i >= PB * PNP * C2) return;
    int c = i % C2;
    int s = (i / C2) % PNP;
    int b = i / (C2 * PNP);
    const float* yb = Y + (size_t)((size_t)b * PNP + s) * PNS * C2 + c;
    float sc = scale[c], sh = shift[c];
    float m = 0.0f;   // relu output is >= 0
    for (int j = 0; j < PNS; ++j)
        m = fmaxf(m, yb[(size_t)j * C2] * sc + sh);
    out[((size_t)b * C2 + c) * PNP + s] = m;
}

// ---------------- host launcher ----------------
extern "C" void kernel_launch(void* const* d_in, const int* in_sizes, int n_in,
                              void* d_out, int out_size, void* d_ws, size_t ws_size,
                              hipStream_t stream)
{
    (void)in_sizes; (void)n_in; (void)out_size; (void)ws_size;

    const float* xyz      = (const float*)d_in[0];
    const float* points   = (const float*)d_in[1];
    const int*   fps_init = (const int*)d_in[2];
    const int*   gidx     = (const int*)d_in[3];
    const float* W0 = (const float*)d_in[4];
    const float* g0 = (const float*)d_in[6];
    const float* be0 = (const float*)d_in[7];
    const float* W1 = (const float*)d_in[8];
    const float* g1 = (const float*)d_in[10];
    const float* be1 = (const float*)d_in[11];
    const float* W2 = (const float*)d_in[12];
    const float* g2 = (const float*)d_in[14];
    const float* be2 = (const float*)d_in[15];
    float* out = (float*)d_out;

    // workspace carving (256B aligned)
    uintptr_t base = (uintptr_t)d_ws;
    size_t off = 0;
    auto carve = [&](size_t bytes) -> void* {
        off = (off + 255) & ~(size_t)255;
        void* p = (void*)(base + off);
        off += bytes;
        return p;
    };
    int*   cents = (int*)  carve((size_t)PB * PNP * sizeof(int));
    float* Wp0   = (float*)carve((size_t)C0 * K0P * sizeof(float));
    float* accS  = (float*)carve(128 * sizeof(float));
    float* accQ  = (float*)carve(128 * sizeof(float));
    float* sc0   = (float*)carve(128 * sizeof(float));
    float* sh0   = (float*)carve(128 * sizeof(float));
    float* sc1   = (float*)carve(128 * sizeof(float));
    float* sh1   = (float*)carve(128 * sizeof(float));
    float* sc2   = (float*)carve(128 * sizeof(float));
    float* sh2   = (float*)carve(128 * sizeof(float));
    float* bufA  = (float*)carve((size_t)PM * 128 * sizeof(float)); // X0 then Y1
    float* bufB  = (float*)carve((size_t)PM * 128 * sizeof(float)); // Y0 then Y2

    const float invM = 1.0f / (float)PM;
    const int gemmBlocks = (PM / 16) * 32 / 256;   // one wave per 16-row strip

    // 1) FPS + new_xyz output
    fps_kernel<<<PB, FPS_THREADS, 0, stream>>>(xyz, fps_init, cents);
    newxyz_kernel<<<(PB * 3 * PNP + 255) / 256, 256, 0, stream>>>(xyz, cents, out);

    // 2) weight pad + gather/concat into X0 (bufA)
    padw_kernel<<<(C0 * K0P + 255) / 256, 256, 0, stream>>>(W0, Wp0);
    {
        long total = (long)PM * K0P;
        gather_kernel<<<(unsigned)((total + 255) / 256), 256, 0, stream>>>(xyz, points, gidx, bufA);
    }

    // 3) layer 0: Y0(bufB) = X0 * W0p^T
    zero_acc<<<1, 128, 0, stream>>>(accS, accQ);
    gemm_wmma<K0P, C0, false><<<gemmBlocks, 256, 0, stream>>>(bufA, Wp0, bufB, nullptr, nullptr);
    stats_kernel<<<2048, 256, 0, stream>>>(bufB, C0, PM / 2048, accS, accQ);
    finalize_stats<<<1, C0, 0, stream>>>(accS, accQ, g0, be0, C0, invM, sc0, sh0);

    // 4) layer 1: Y1(bufA) = relu(bn(Y0)) * W1^T  (norm fused into A-loads)
    zero_acc<<<1, 128, 0, stream>>>(accS, accQ);
    gemm_wmma<C0, C1, true><<<gemmBlocks, 256, 0, stream>>>(bufB, W1, bufA, sc0, sh0);
    stats_kernel<<<2048, 256, 0, stream>>>(bufA, C1, PM / 2048, accS, accQ);
    finalize_stats<<<1, C1, 0, stream>>>(accS, accQ, g1, be1, C1, invM, sc1, sh1);

    // 5) layer 2: Y2(bufB) = relu(bn(Y1)) * W2^T
    zero_acc<<<1, 128, 0, stream>>>(accS, accQ);
    gemm_wmma<C1, C2, true><<<gemmBlocks, 256, 0, stream>>>(bufA, W2, bufB, sc1, sh1);
    stats_kernel<<<2048, 256, 0, stream>>>(bufB, C2, PM / 2048, accS, accQ);
    finalize_stats<<<1, C2, 0, stream>>>(accS, accQ, g2, be2, C2, invM, sc2, sh2);

    // 6) fused bn+relu+max over NS -> new_points [B,128,NP]
    maxpool_kernel<<<(PB * PNP * C2 + 255) / 256, 256, 0, stream>>>(
        bufB, sc2, sh2, out + (size_t)PB * 3 * PNP);
}